// V23Model_35854386987257
// MI455X (gfx1250) — compile-verified
//
#include <hip/hip_runtime.h>
#include <math.h>

typedef __attribute__((ext_vector_type(2))) float v2f;
typedef __attribute__((ext_vector_type(8))) float v8f;

#define LRES 4096
#define NTI (LRES / 16)                    // 256 i-tiles
#define NCHUNK 16                          // j-chunks per row strip
#define TILES_PER_CHUNK (NTI / NCHUNK)     // 16 j-tiles per wave sweep
#define WAVES_PER_BLOCK 8
#define PAIR_BLOCKS ((NTI * NCHUNK) / WAVES_PER_BLOCK)  // 512

// ---- workspace layout (float offsets) ----
enum : int {
  OFF_CA4 = 0,
  OFF_N4  = OFF_CA4 + 4 * LRES,
  OFF_O4  = OFF_N4  + 4 * LRES,
  OFF_CB4 = OFF_O4  + 4 * LRES,
  OFF_CO4 = OFF_CB4 + 4 * LRES,
  OFF_CA2 = OFF_CO4 + 4 * LRES,
  OFF_O2  = OFF_CA2 + LRES,
  OFF_N2  = OFF_O2  + LRES,
  OFF_CB2 = OFF_N2  + LRES,
  OFF_Q   = OFF_CB2 + LRES,
  OFF_HYD = OFF_Q   + LRES,
  OFF_COLEN  = OFF_HYD + LRES,
  OFF_CODOTO = OFF_COLEN + LRES,
  OFF_IDEAL  = OFF_CODOTO + LRES,
  OFF_DENS   = OFF_IDEAL + LRES,
  OFF_DMIN   = OFF_DENS + LRES,    // float bits used as ordered uint
  OFF_ACC    = OFF_DMIN + LRES,
  WS_FLOATS  = OFF_ACC + 16
};
enum : int { ACC_BOND = 0, ACC_ANGLE, ACC_RAMA, ACC_ENT, ACC_SMOOTH, ACC_ELEC, ACC_HB };

// ---- per-residue tables ('ACDEFGHIKLMNPQRSTVWYX') ----
__constant__ float c_charge[21] = {0,0,-1,-1,0,0,0.5f,0,1,0,0,0,0,0,1,0,0,0,0,0,0};
__constant__ float c_hydro[21]  = {1.8f,2.5f,-3.5f,-3.5f,2.8f,-0.4f,-3.2f,4.5f,-3.9f,3.8f,
                                   1.9f,-3.5f,-1.6f,-3.5f,-4.5f,-0.8f,-0.7f,4.2f,-0.9f,-1.3f,0.0f};
__constant__ float c_phi0[21] = {-60,-60,-60,-60,-60,-75,-60,-60,-60,-60,-60,-60,-65,-60,-60,-60,-60,-60,-60,-60,-60};
__constant__ float c_psi0[21] = {-45,-45,-45,-45,-45,-60,-45,-45,-45,-45,-45,-45,-30,-45,-45,-45,-45,-45,-45,-45,-45};
__constant__ float c_wid[21]  = {25,25,25,25,25,40,25,25,25,25,25,25,20,25,25,25,25,25,25,25,25};

// ---- small float3 helpers ----
struct F3 { float x, y, z; };
__device__ inline F3 mk3(float x, float y, float z) { return F3{x, y, z}; }
__device__ inline F3 add3(F3 a, F3 b) { return F3{a.x + b.x, a.y + b.y, a.z + b.z}; }
__device__ inline F3 sub3(F3 a, F3 b) { return F3{a.x - b.x, a.y - b.y, a.z - b.z}; }
__device__ inline F3 scl3(F3 a, float s) { return F3{a.x * s, a.y * s, a.z * s}; }
__device__ inline float dot3(F3 a, F3 b) { return a.x * b.x + a.y * b.y + a.z * b.z; }
__device__ inline F3 crs3(F3 a, F3 b) {
  return F3{a.y * b.z - a.z * b.y, a.z * b.x - a.x * b.z, a.x * b.y - a.y * b.x};
}
__device__ inline F3 nrm3(F3 v) {   // matches _normalize: x / max(|x|, 1e-8)
  float n = sqrtf(dot3(v, v));
  float s = 1.0f / fmaxf(n, 1e-8f);
  return scl3(v, s);
}
__device__ inline F3 ld3(const float* p, int i) { return F3{p[3*i], p[3*i+1], p[3*i+2]}; }
__device__ inline void st4(float* base, int i, F3 v) {
  base[4*i] = v.x; base[4*i+1] = v.y; base[4*i+2] = v.z; base[4*i+3] = 0.0f;
}

__device__ inline float dihedral(F3 p0, F3 p1, F3 p2, F3 p3) {
  F3 b0 = sub3(p1, p0), b1 = sub3(p2, p1), b2 = sub3(p3, p2);
  F3 b1n = nrm3(b1);
  F3 v = sub3(b0, scl3(b1n, dot3(b0, b1n)));
  F3 w = sub3(b2, scl3(b1n, dot3(b2, b1n)));
  float x = dot3(v, w);
  float y = dot3(crs3(b1n, v), w);
  return atan2f(y + 1e-8f, x + 1e-8f);
}

__device__ inline float redSum32(float v) {
  for (int m = 16; m >= 1; m >>= 1) v += __shfl_xor(v, m, 32);
  return v;
}
__device__ inline float redSum16(float v) {
  for (int m = 8; m >= 1; m >>= 1) v += __shfl_xor(v, m, 16);
  return v;
}
__device__ inline float redMin16(float v) {
  for (int m = 8; m >= 1; m >>= 1) v = fminf(v, __shfl_xor(v, m, 16));
  return v;
}

// ============ kernel 0: init accumulators ============
__global__ void k_init(float* ws) {
  int i = blockIdx.x * blockDim.x + threadIdx.x;
  if (i < LRES) {
    ws[OFF_DENS + i] = 0.0f;
    ((unsigned*)ws)[OFF_DMIN + i] = __float_as_uint(1e30f);
  }
  if (i < 16) ws[OFF_ACC + i] = 0.0f;
}

// ============ kernel 1: per-residue precompute + O(L) energies ============
__global__ __launch_bounds__(256) void k_precompute(const float* __restrict__ ca,
                                                    const float* __restrict__ alpha,
                                                    const int* __restrict__ seq,
                                                    float* __restrict__ ws) {
  int i = blockIdx.x * blockDim.x + threadIdx.x;
  float cb_bond = 0.f, cb_ang = 0.f, cb_rama = 0.f, cb_ent = 0.f, cb_sm = 0.f;
  if (i < LRES) {
    F3 p  = ld3(ca, i);
    F3 pm = (i > 0)        ? ld3(ca, i - 1) : p;
    F3 pp = (i < LRES - 1) ? ld3(ca, i + 1) : p;
    F3 vn_prev = mk3(0, 0, 0), vn_next = mk3(0, 0, 0);
    if (i > 0)        vn_prev = nrm3(sub3(p, pm));
    if (i < LRES - 1) vn_next = nrm3(sub3(pp, p));

    F3 N = (i == 0)       ? sub3(p, scl3(vn_next, 1.45f)) : sub3(p, scl3(vn_prev, 1.45f));
    F3 C = (i < LRES - 1) ? add3(p, scl3(vn_next, 1.52f)) : add3(p, scl3(vn_prev, 1.52f));

    F3 perp = crs3(sub3(C, p), sub3(N, p));
    float pn = sqrtf(dot3(perp, perp));
    if (pn > 1e-6f) perp = scl3(perp, 1.0f / pn);
    F3 O = (i < LRES - 1) ? add3(C, scl3(perp, 1.24f)) : mk3(C.x, C.y + 1.24f, C.z);

    F3 co = sub3(O, C);
    F3 vcb = scl3(add3(sub3(N, p), sub3(C, p)), -1.0f);
    F3 CB = add3(p, scl3(nrm3(vcb), 1.8f));

    st4(ws + OFF_CA4, i, p);
    st4(ws + OFF_N4,  i, N);
    st4(ws + OFF_O4,  i, O);
    st4(ws + OFF_CB4, i, CB);
    st4(ws + OFF_CO4, i, co);
    ws[OFF_CA2 + i] = dot3(p, p);
    ws[OFF_O2  + i] = dot3(O, O);
    ws[OFF_N2  + i] = dot3(N, N);
    ws[OFF_CB2 + i] = dot3(CB, CB);
    int s = seq[i]; s = (s < 0) ? 0 : (s > 20 ? 20 : s);
    ws[OFF_Q   + i] = c_charge[s];
    ws[OFF_HYD + i] = c_hydro[s];
    ws[OFF_COLEN  + i] = fmaxf(sqrtf(dot3(co, co)), 1e-8f);
    ws[OFF_CODOTO + i] = dot3(co, O);
    float ai = alpha[i];
    ws[OFF_IDEAL + i] = 2.9f * (1.0f + 0.1f * (ai - 1.0f));

    if (i < LRES - 1) {  // bond + alpha smoothness
      F3 dv = sub3(pp, p);
      float d = sqrtf(dot3(dv, dv));
      float t0 = 3.8f * (1.0f + 0.1f * (ai - 1.0f));
      float t1 = 3.8f * (1.0f + 0.1f * (alpha[i + 1] - 1.0f));
      float dd = d - 0.5f * (t0 + t1);
      cb_bond = dd * dd;
      float da = alpha[i + 1] - ai;
      cb_sm = da * da;
    }
    if (i > 0 && i < LRES - 1) {  // angle + rama
      F3 v1 = nrm3(sub3(pm, p)), v2 = nrm3(sub3(pp, p));
      float ct = cosf(1.5708f * (1.0f + 0.1f * (ai - 1.0f)));
      float dc = dot3(v1, v2) - ct;
      cb_ang = dc * dc;
      F3 Cprev = add3(pm, scl3(vn_prev, 1.52f));
      F3 Nnext = sub3(pp, scl3(vn_next, 1.45f));
      const float deg = 57.29577951308232f;
      float phi = dihedral(Cprev, N, p, C) * deg;
      float psi = dihedral(N, p, C, Nnext) * deg;
      float weff = c_wid[s] * (1.0f + 0.1f * (ai - 1.0f)) + 1e-8f;
      float dphi = (phi - c_phi0[s]) / weff;
      float dpsi = (psi - c_psi0[s]) / weff;
      cb_rama = dphi * dphi + dpsi * dpsi;
    }
    cb_ent = ai * __logf(ai + 1e-8f);
  }
  cb_bond = redSum32(cb_bond);
  cb_ang  = redSum32(cb_ang);
  cb_rama = redSum32(cb_rama);
  cb_ent  = redSum32(cb_ent);
  cb_sm   = redSum32(cb_sm);
  if ((threadIdx.x & 31) == 0) {
    atomicAdd(ws + OFF_ACC + ACC_BOND,   cb_bond);
    atomicAdd(ws + OFF_ACC + ACC_ANGLE,  cb_ang);
    atomicAdd(ws + OFF_ACC + ACC_RAMA,   cb_rama);
    atomicAdd(ws + OFF_ACC + ACC_ENT,    cb_ent);
    atomicAdd(ws + OFF_ACC + ACC_SMOOTH, cb_sm);
  }
}

// ============ kernel 2: fused O(L^2) pair sweep via V_WMMA_F32_16X16X4_F32 ============
__global__ __launch_bounds__(256) void k_pairs(float* __restrict__ ws) {
  const float* caP = ws + OFF_CA4;
  const float* nP  = ws + OFF_N4;
  const float* oP  = ws + OFF_O4;
  const float* cbP = ws + OFF_CB4;
  const float* coP = ws + OFF_CO4;
  unsigned* dminb = ((unsigned*)ws) + OFF_DMIN;

  const int lane = threadIdx.x & 31;
  const int wid  = threadIdx.x >> 5;
  const int gw   = blockIdx.x * WAVES_PER_BLOCK + wid;
  const int ti    = gw / NCHUNK;
  const int chunk = gw % NCHUNK;
  const int i0 = ti * 16;
  const int m = lane & 15;          // A row / B col within tile
  const int half = lane >> 4;       // selects K pair {2h, 2h+1}

  // A fragments: lane holds components (2h, 2h+1) of strip row m (float4 pad, w=0)
  v2f aCA = *(const v2f*)(caP + (i0 + m) * 4 + 2 * half);
  v2f aO  = *(const v2f*)(oP  + (i0 + m) * 4 + 2 * half);
  v2f aCB = *(const v2f*)(cbP + (i0 + m) * 4 + 2 * half);
  v2f aCO = *(const v2f*)(coP + (i0 + m) * 4 + 2 * half);

  float row_ca2[8], row_o2[8], row_cb2[8], row_q[8], row_colen[8], row_codoto[8], row_ideal[8];
  float dens_loc[8], dmin2_loc[8];
#pragma unroll
  for (int r = 0; r < 8; r++) {
    int i = i0 + r + 8 * half;       // C/D row for accumulator element r
    row_ca2[r]    = ws[OFF_CA2 + i];
    row_o2[r]     = ws[OFF_O2 + i];
    row_cb2[r]    = ws[OFF_CB2 + i];
    row_q[r]      = ws[OFF_Q + i];
    row_colen[r]  = ws[OFF_COLEN + i];
    row_codoto[r] = ws[OFF_CODOTO + i];
    row_ideal[r]  = ws[OFF_IDEAL + i];
    dens_loc[r]  = 0.0f;
    dmin2_loc[r] = 1e30f;            // min of squared distances (sqrt deferred)
  }
  float sum_e = 0.0f, sum_h = 0.0f;

  for (int jt = 0; jt < TILES_PER_CHUNK; ++jt) {
    const int j0 = chunk * (TILES_PER_CHUNK * 16) + jt * 16;
    v2f bCA = *(const v2f*)(caP + (j0 + m) * 4 + 2 * half);
    v2f bN  = *(const v2f*)(nP  + (j0 + m) * 4 + 2 * half);
    const int j = j0 + m;
    const float c_ca2 = ws[OFF_CA2 + j];
    const float c_n2  = ws[OFF_N2 + j];
    const float c_q   = ws[OFF_Q + j];

    v8f z = {0.f, 0.f, 0.f, 0.f, 0.f, 0.f, 0.f, 0.f};
    // One K=4 WMMA completes a full 3D-dot 16x16 tile (w component is 0).
    v8f dCA = __builtin_amdgcn_wmma_f32_16x16x4_f32(false, aCA, false, bCA, (short)0, z, false, false);
    v8f dON = __builtin_amdgcn_wmma_f32_16x16x4_f32(false, aO,  false, bN,  (short)0, z, false, false);
    v8f dCB = __builtin_amdgcn_wmma_f32_16x16x4_f32(false, aCB, false, bCA, (short)0, z, false, false);
    v8f dCO = __builtin_amdgcn_wmma_f32_16x16x4_f32(false, aCO, false, bN,  (short)0, z, false, false);

#pragma unroll
    for (int r = 0; r < 8; r++) {
      const int i = i0 + r + 8 * half;
      const bool diag = (i == j);
      // electro + density (D = cdist(ca,ca) in Gram form, like the reference)
      float d2 = row_ca2[r] + c_ca2 - 2.0f * dCA[r];
      float D  = __builtin_amdgcn_sqrtf(fmaxf(d2, 0.0f));   // raw v_sqrt_f32
      float De = D + 1e-6f;
      float Ee = row_q[r] * c_q * __expf(-0.1f * De) * __builtin_amdgcn_rcpf(80.0f * De);
      sum_e += diag ? 0.0f : Ee;
      dens_loc[r] += (d2 < 100.0f) ? 1.0f : 0.0f;           // D<10 on squared value
      // hbond: Dh = |O_i - N_j|; align = co_i.(N_j - O_i) / (|co_i| |N_j - O_i|)
      float Dh = __builtin_amdgcn_sqrtf(fmaxf(row_o2[r] + c_n2 - 2.0f * dON[r], 0.0f));
      float align = (dCO[r] - row_codoto[r]) *
                    __builtin_amdgcn_rcpf(row_colen[r] * fmaxf(Dh, 1e-8f));
      float t  = (Dh - row_ideal[r]) * (1.0f / 0.3f);
      float Eh = -align * __expf(-t * t);
      sum_h += ((Dh > 2.5f) && (Dh < 3.5f)) ? Eh : 0.0f;
      // rotamer: track min squared distance; sqrt deferred (monotonic).
      // Reference's diagonal +1e9 can never be the row min, so push it out here.
      float dr2 = fmaxf(row_cb2[r] + c_ca2 - 2.0f * dCB[r], 0.0f);
      dr2 = diag ? 1e30f : dr2;
      dmin2_loc[r] = fminf(dmin2_loc[r], dr2);
    }
  }

  // per-row reductions across the 16 column-lanes of each half, 1 atomic per row per wave
#pragma unroll
  for (int r = 0; r < 8; r++) {
    float d  = redSum16(dens_loc[r]);
    float mn = __builtin_amdgcn_sqrtf(redMin16(dmin2_loc[r]));  // one sqrt per row
    if (m == 0) {
      int i = i0 + r + 8 * half;
      atomicAdd(ws + OFF_DENS + i, d);
      atomicMin(dminb + i, __float_as_uint(mn));   // nonneg floats: uint order == float order
    }
  }
  sum_e = redSum32(sum_e);
  sum_h = redSum32(sum_h);
  if (lane == 0) {
    atomicAdd(ws + OFF_ACC + ACC_ELEC, sum_e);
    atomicAdd(ws + OFF_ACC + ACC_HB,   sum_h);
  }
}

// ============ kernel 3: solvent + rotamer + weighted combine ============
__global__ __launch_bounds__(256) void k_finalize(const int* __restrict__ seq,
                                                  const float* __restrict__ ws,
                                                  float* __restrict__ out) {
  __shared__ float s1[256], s2[256];
  float sum_solv = 0.0f, sum_rot = 0.0f;
  for (int i = threadIdx.x; i < LRES; i += 256) {
    float dens = ws[OFF_DENS + i];
    float burial = 1.0f - __expf(-dens * 0.05f);
    float h = ws[OFF_HYD + i];
    sum_solv += (h > 0.0f) ? h * (1.0f - burial) : (-h * burial);
    float dm = __uint_as_float(((const unsigned*)ws)[OFF_DMIN + i]);
    bool inc = (seq[i] != 5) && (i > 0) && (i < LRES - 1);
    if (inc) sum_rot += fmaxf(4.0f - dm, 0.0f);
  }
  s1[threadIdx.x] = sum_solv;
  s2[threadIdx.x] = sum_rot;
  __syncthreads();
  for (int st = 128; st > 0; st >>= 1) {
    if (threadIdx.x < st) {
      s1[threadIdx.x] += s1[threadIdx.x + st];
      s2[threadIdx.x] += s2[threadIdx.x + st];
    }
    __syncthreads();
  }
  if (threadIdx.x == 0) {
    const float LL = (float)LRES * (float)LRES;
    float e = 0.0f;
    e += 1.0f  * ws[OFF_ACC + ACC_BOND]  / (float)(LRES - 1);   // W_BOND
    e += 1.0f  * ws[OFF_ACC + ACC_ANGLE] / (float)(LRES - 2);   // W_ANGLE
    e += 0.5f  * ws[OFF_ACC + ACC_RAMA]  / (float)(LRES - 2);   // W_RAMA
    // clash term is identically zero in the reference
    e += 1.0f  * ws[OFF_ACC + ACC_HB]   / LL;                   // W_HBOND
    e += 0.5f  * ws[OFF_ACC + ACC_ELEC] / LL;                   // W_ELECTRO
    e += 0.5f  * s1[0] / (float)LRES;                           // W_SOLVENT
    e += 0.5f  * s2[0] / (float)(LRES - 2);                     // W_ROTAMER
    e += 0.01f * (-ws[OFF_ACC + ACC_ENT] / (float)LRES)         // W_AENT
       + 0.01f * ws[OFF_ACC + ACC_SMOOTH] / (float)(LRES - 1);  // W_ASMOOTH
    out[0] = e;
  }
}

extern "C" void kernel_launch(void* const* d_in, const int* in_sizes, int n_in,
                              void* d_out, int out_size, void* d_ws, size_t ws_size,
                              hipStream_t stream) {
  const float* ca    = (const float*)d_in[0];  // [4096,3]
  const float* alpha = (const float*)d_in[1];  // [4096]
  const int*   seq   = (const int*)d_in[2];    // [4096]
  float* ws  = (float*)d_ws;                   // needs WS_FLOATS*4 ≈ 508 KB
  float* out = (float*)d_out;

  k_init<<<LRES / 256, 256, 0, stream>>>(ws);
  k_precompute<<<LRES / 256, 256, 0, stream>>>(ca, alpha, seq, ws);
  k_pairs<<<PAIR_BLOCKS, 256, 0, stream>>>(ws);
  k_finalize<<<1, 256, 0, stream>>>(seq, ws, out);
}